// PostProcessor_12163347382878
// MI455X (gfx1250) — compile-verified
//
#include <hip/hip_runtime.h>
#include <hip/hip_bf16.h>
#include <math.h>
#include <stdint.h>

// Problem constants (match reference)
#define B_IMG   2
#define N_BOX   1000
#define N_CLS   81
#define N_FG    80          // foreground classes
#define PAD     1024        // sort padding (next pow2 >= N_BOX)
#define DET     100
#define SCORE_TH 0.05f
#define NMS_TH   0.5f
#define BBOX_CLIP 4.135166556742356f   // log(1000/16)

// ---------------------------------------------------------------------------
// Kernel 1: softmax + box decode + clip.
// One wave32 per proposal row. 256 threads = 8 rows per block.
// ---------------------------------------------------------------------------
__global__ __launch_bounds__(256)
void decode_kernel(const float* __restrict__ logits,      // [B*N, 81]
                   const float* __restrict__ regr,        // [B*N, 324]
                   const float* __restrict__ props,       // [B, N, 4]
                   const int*   __restrict__ imgsz,       // [B, 2] (W,H)
                   float* __restrict__ scores,            // [B, 80, N]
                   float* __restrict__ boxes)             // [B, 80, N, 4]
{
    const int lane = threadIdx.x & 31;
    const int wv   = threadIdx.x >> 5;
    const int r    = blockIdx.x * 8 + wv;
    if (r >= B_IMG * N_BOX) return;
    const int b = r / N_BOX;
    const int n = r % N_BOX;

    const float* x = logits + (size_t)r * N_CLS;

    // cache this lane's logits (classes lane, lane+32, lane+64)
    const float l0 = x[lane];
    const float l1 = (lane + 32 < N_CLS) ? x[lane + 32] : -__builtin_inff();
    const float l2 = (lane + 64 < N_CLS) ? x[lane + 64] : -__builtin_inff();

    // wave-wide max
    float m = fmaxf(l0, fmaxf(l1, l2));
    #pragma unroll
    for (int o = 16; o > 0; o >>= 1) m = fmaxf(m, __shfl_xor(m, o, 32));
    // wave-wide sum of exp
    float s = expf(l0 - m);
    if (lane + 32 < N_CLS) s += expf(l1 - m);
    if (lane + 64 < N_CLS) s += expf(l2 - m);
    #pragma unroll
    for (int o = 16; o > 0; o >>= 1) s += __shfl_xor(s, o, 32);

    // proposal geometry (uniform within the wave's row)
    const float* pb = props + (size_t)r * 4;
    const float px1 = pb[0], py1 = pb[1], px2 = pb[2], py2 = pb[3];
    const float wdt = px2 - px1 + 1.0f;
    const float hgt = py2 - py1 + 1.0f;
    const float cx  = px1 + 0.5f * wdt;
    const float cy  = py1 + 0.5f * hgt;
    const float xmax = (float)imgsz[b * 2 + 0] - 1.0f;
    const float ymax = (float)imgsz[b * 2 + 1] - 1.0f;

    const float* rg = regr + (size_t)r * (N_CLS * 4);

    for (int c = 1 + lane; c < N_CLS; c += 32) {
        const float p = expf(x[c] - m) / s;
        const size_t o1 = (size_t)(b * N_FG + (c - 1)) * N_BOX + n;
        scores[o1] = p;

        const float dx = rg[c * 4 + 0] / 10.0f;
        const float dy = rg[c * 4 + 1] / 10.0f;
        const float dw = fminf(rg[c * 4 + 2] / 5.0f, BBOX_CLIP);
        const float dh = fminf(rg[c * 4 + 3] / 5.0f, BBOX_CLIP);
        const float pcx = dx * wdt + cx;
        const float pcy = dy * hgt + cy;
        const float pw  = expf(dw) * wdt;
        const float ph  = expf(dh) * hgt;
        float bx1 = pcx - 0.5f * pw;
        float by1 = pcy - 0.5f * ph;
        float bx2 = pcx + 0.5f * pw - 1.0f;
        float by2 = pcy + 0.5f * ph - 1.0f;
        bx1 = fminf(fmaxf(bx1, 0.0f), xmax);
        by1 = fminf(fmaxf(by1, 0.0f), ymax);
        bx2 = fminf(fmaxf(bx2, 0.0f), xmax);
        by2 = fminf(fmaxf(by2, 0.0f), ymax);
        float* bo = boxes + o1 * 4;
        bo[0] = bx1; bo[1] = by1; bo[2] = bx2; bo[3] = by2;
    }
}

// ---------------------------------------------------------------------------
// Kernel 2: exact per-class NMS. One 256-thread block per (image,class).
// Empty classes early-exit; bitonic sort (desc by score) of 1024 packed keys
// in LDS; sorted valid-prefix boxes async-gathered into LDS via
// global_load_async_to_lds_b128 (ASYNCcnt); sequential greedy suppression.
// ---------------------------------------------------------------------------
__global__ __launch_bounds__(256)
void nms_kernel(const float* __restrict__ scores,   // [B, 80, N]
                const float* __restrict__ boxes,    // [B, 80, N, 4]
                float* __restrict__ masked)         // [B, 80, N]
{
    const int blk = blockIdx.x;
    const int b   = blk / N_FG;
    const int c   = blk % N_FG;
    const float* sc = scores + (size_t)(b * N_FG + c) * N_BOX;
    const float* bx = boxes  + (size_t)(b * N_FG + c) * N_BOX * 4;
    float*       mk = masked + (size_t)(b * N_FG + c) * N_BOX;

    __shared__ unsigned long long key[PAD];      // (desc-score-bits << 32) | idx
    __shared__ __align__(16) float sbox[PAD * 4];
    __shared__ float sarea[PAD];
    __shared__ float sscore[PAD];
    __shared__ int   skeep[PAD];
    __shared__ int   scount;

    const int tid = threadIdx.x;
    const int TB  = 256;

    if (tid == 0) scount = 0;
    __syncthreads();

    // Phase 1: build sort keys (descending score via order-flip transform).
    for (int k = tid; k < PAD; k += TB) {
        float sv = (k < N_BOX) ? sc[k] : -__builtin_inff();
        const bool valid = (k < N_BOX) && (sv > SCORE_TH);
        if (valid) atomicAdd(&scount, 1);
        const float kf = valid ? sv : -__builtin_inff();
        unsigned u   = __float_as_uint(kf);
        unsigned asc = (u & 0x80000000u) ? ~u : (u | 0x80000000u);
        unsigned dsc = ~asc;                 // ascending sort of dsc == desc score
        key[k] = ((unsigned long long)dsc << 32) | (unsigned)k;
    }
    __syncthreads();
    const int nvalid = scount;

    // Common case: nothing above threshold for this class -> just -1 fill.
    if (nvalid == 0) {
        for (int t = tid; t < N_BOX; t += TB) mk[t] = -1.0f;
        return;
    }

    // Pre-warm the box slice we will gather later (gfx1250 global_prefetch_b8).
    for (int t = tid * 32; t < N_BOX * 4; t += TB * 32)
        __builtin_prefetch(bx + t, 0, 0);

    // Phase 2: bitonic sort ascending over PAD keys (valid prefix first).
    for (int kk = 2; kk <= PAD; kk <<= 1) {
        for (int j = kk >> 1; j > 0; j >>= 1) {
            for (int i = tid; i < PAD; i += TB) {
                const int ixj = i ^ j;
                if (ixj > i) {
                    const bool up = ((i & kk) == 0);
                    const unsigned long long a = key[i];
                    const unsigned long long d = key[ixj];
                    if (up ? (a > d) : (a < d)) { key[i] = d; key[ixj] = a; }
                }
            }
            __syncthreads();
        }
    }

    // Phase 3: gather scores + boxes for the sorted valid prefix only.
    // Boxes go straight to LDS with CDNA5 async global->LDS b128 (ASYNCcnt).
    for (int k = tid; k < nvalid; k += TB) {
        const unsigned n = (unsigned)(key[k] & 0xFFFFFFFFu);
        sscore[k] = sc[n];
        skeep[k]  = 1;
#if defined(__gfx1250__)
        // flat LDS pointer: low 32 bits are the workgroup-relative LDS offset
        const unsigned lds_off = (unsigned)(uintptr_t)(&sbox[4 * k]);
        const unsigned long long ga = (unsigned long long)(uintptr_t)(bx + 4 * (size_t)n);
        asm volatile("global_load_async_to_lds_b128 %0, %1, off"
                     :: "v"(lds_off), "v"(ga) : "memory");
#else
        const float4 v = *(const float4*)(bx + 4 * (size_t)n);
        ((float4*)sbox)[k] = v;
#endif
    }
#if defined(__gfx1250__)
    asm volatile("s_wait_asynccnt 0" ::: "memory");
#endif
    __syncthreads();

    // Phase 4: areas (reference: no +1, clamp at 0).
    for (int k = tid; k < nvalid; k += TB) {
        const float w = fmaxf(sbox[4 * k + 2] - sbox[4 * k + 0], 0.0f);
        const float h = fmaxf(sbox[4 * k + 3] - sbox[4 * k + 1], 0.0f);
        sarea[k] = w * h;
    }
    __syncthreads();

    // Phase 5: sequential greedy suppression over the valid prefix.
    for (int i = 0; i < nvalid; ++i) {
        if (!skeep[i]) continue;             // uniform branch (LDS, post-barrier)
        const float x1i = sbox[4 * i + 0], y1i = sbox[4 * i + 1];
        const float x2i = sbox[4 * i + 2], y2i = sbox[4 * i + 3];
        const float ai  = sarea[i];
        for (int j = i + 1 + tid; j < nvalid; j += TB) {
            if (skeep[j]) {
                const float xx1 = fmaxf(x1i, sbox[4 * j + 0]);
                const float yy1 = fmaxf(y1i, sbox[4 * j + 1]);
                const float xx2 = fminf(x2i, sbox[4 * j + 2]);
                const float yy2 = fminf(y2i, sbox[4 * j + 3]);
                const float w = fmaxf(xx2 - xx1, 0.0f);
                const float h = fmaxf(yy2 - yy1, 0.0f);
                const float inter = w * h;
                const float iou = inter / (ai + sarea[j] - inter + 1e-12f);
                if (iou > NMS_TH) skeep[j] = 0;
            }
        }
        __syncthreads();
    }

    // Phase 6: masked scores at original indices (-1 where not kept).
    for (int t = tid; t < N_BOX; t += TB) mk[t] = -1.0f;
    __threadfence_block();
    __syncthreads();
    for (int k = tid; k < nvalid; k += TB) {
        if (skeep[k]) {
            const unsigned n = (unsigned)(key[k] & 0xFFFFFFFFu);
            mk[n] = sscore[k];
        }
    }
}

// ---------------------------------------------------------------------------
// Kernel 3: per-image top-100 over [80*N] masked scores + output assembly.
// Early-break once only -1 sentinels remain (output-identical to reference).
// ---------------------------------------------------------------------------
__global__ __launch_bounds__(1024)
void topk_kernel(float* __restrict__ masked,        // [B, 80, N] (mutated)
                 const float* __restrict__ boxes,   // [B, 80, N, 4]
                 float* __restrict__ out)           // [B, 100, 6]
{
    const int b = blockIdx.x;
    float* M = masked + (size_t)b * N_FG * N_BOX;
    const int TOT = N_FG * N_BOX;
    const int tid = threadIdx.x;

    __shared__ float rv[1024];
    __shared__ int   ri[1024];
    __shared__ float selv[DET];
    __shared__ int   seli[DET];

    for (int d = 0; d < DET; ++d) {
        float bv = -__builtin_inff();
        int   bi = TOT;
        for (int e = tid; e < TOT; e += 1024) {
            const float v = M[e];
            if (v > bv || (v == bv && e < bi)) { bv = v; bi = e; }
        }
        rv[tid] = bv; ri[tid] = bi;
        __syncthreads();
        #pragma unroll
        for (int off = 512; off > 0; off >>= 1) {
            if (tid < off) {
                const float v2 = rv[tid + off];
                const int   i2 = ri[tid + off];
                if (v2 > rv[tid] || (v2 == rv[tid] && i2 < ri[tid])) {
                    rv[tid] = v2; ri[tid] = i2;
                }
            }
            __syncthreads();
        }
        const float top = rv[0];           // uniform (post-barrier)
        if (top <= 0.0f) {
            // Only -1 sentinels remain: which index wins is irrelevant to the
            // output (rows are zeroed / label -1). Fill remainder and stop.
            for (int t = d + tid; t < DET; t += 1024) { selv[t] = -1.0f; seli[t] = 0; }
            __syncthreads();
            break;
        }
        if (tid == 0) {
            selv[d] = top;
            seli[d] = ri[0];
            M[ri[0]] = -2.0f;              // consume winner
        }
        __threadfence_block();
        __syncthreads();
    }

    if (tid < DET) {
        const float v = selv[tid];
        const int   i = seli[tid];
        const bool ok = v > 0.0f;
        const int c0  = i / N_BOX;
        const int n   = i % N_BOX;
        const float* bp = boxes + ((size_t)(b * N_FG + c0) * N_BOX + n) * 4;
        float* o = out + (size_t)(b * DET + tid) * 6;
        o[0] = ok ? bp[0] : 0.0f;
        o[1] = ok ? bp[1] : 0.0f;
        o[2] = ok ? bp[2] : 0.0f;
        o[3] = ok ? bp[3] : 0.0f;
        o[4] = ok ? v : 0.0f;
        o[5] = ok ? (float)(c0 + 1) : -1.0f;
    }
}

// ---------------------------------------------------------------------------
extern "C" void kernel_launch(void* const* d_in, const int* in_sizes, int n_in,
                              void* d_out, int out_size, void* d_ws, size_t ws_size,
                              hipStream_t stream) {
    (void)in_sizes; (void)n_in; (void)out_size; (void)ws_size;
    const float* class_logits   = (const float*)d_in[0];  // [2000, 81]
    const float* box_regression = (const float*)d_in[1];  // [2000, 324]
    const float* proposal_boxes = (const float*)d_in[2];  // [2, 1000, 4]
    const int*   image_sizes    = (const int*)  d_in[3];  // [2, 2]

    float* ws        = (float*)d_ws;
    float* ws_scores = ws;                                   // 160000 f32
    float* ws_boxes  = ws + (size_t)B_IMG * N_FG * N_BOX;    // 640000 f32
    float* ws_masked = ws_boxes + (size_t)B_IMG * N_FG * N_BOX * 4; // 160000 f32

    decode_kernel<<<(B_IMG * N_BOX + 7) / 8, 256, 0, stream>>>(
        class_logits, box_regression, proposal_boxes, image_sizes,
        ws_scores, ws_boxes);

    nms_kernel<<<B_IMG * N_FG, 256, 0, stream>>>(ws_scores, ws_boxes, ws_masked);

    topk_kernel<<<B_IMG, 1024, 0, stream>>>(ws_masked, ws_boxes, (float*)d_out);
}